// junsteak_30081950941495
// MI455X (gfx1250) — compile-verified
//
#include <hip/hip_runtime.h>

// Slab-ocean Euler recurrence recast as affine-operator algebra on MI455X.
//   S_{k+1} = M S_k + DT*Kin0*(TAxnow e0 + TAynow e32),  M = I + DT*J (64x64, f32)
//   => per forcing interval n: S_{n+1} = M^60 S_n + c_n
//   => with S_0 = 0 the outer scan is a prefix-convolution with powers of M^60:
//      chunked (45 x 16) into parallel local prefixes + 45-step boundary scan +
//      WMMA batched fix-up  S_{16g+r} = (M^60)^r B_g + l_r.
// Precision: M ~ I + O(1e-3) -> f32 WMMA (V_WMMA_F32_16X16X4_F32) is required;
// f16/bf16 mantissas would destroy the perturbation under 60-fold composition.
// Latency model: everything fits in LDS/L2; HBM traffic ~ outputs (184 KB,
// ~8 ns at 23.3 TB/s). Runtime is sequential-depth bound, so depth is cut from
// 43,200 Euler steps -> 59 (power chain) + 16 + 45 matvec-equivalents.

#define NL 32
#define NST 64
#define NSUB 60
#define DT 60.0f
#define CHUNK 16
#define BPAD 48          // 45 chunk columns padded to 3 WMMA tiles

typedef __attribute__((ext_vector_type(2))) float v2f;
typedef __attribute__((ext_vector_type(8))) float v8f;

// ---------------- K1: build M = I + DT*J, plus P[0]=e0, Q[0]=e32 ----------------
__global__ void __launch_bounds__(256) junsteak_build_M(
    const float* __restrict__ pk, const float* __restrict__ fc,
    float* __restrict__ M, float* __restrict__ P, float* __restrict__ Q) {
  const float f = fc[0];
  for (int idx = threadIdx.x; idx < NST * NST; idx += blockDim.x) {
    const int i = idx >> 6, j = idx & 63;
    const int bi = i >> 5, bj = j >> 5;   // 0 = U block, 1 = V block
    const int r = i & 31, c = j & 31;
    float val = 0.0f;
    if (bi == bj) {
      const float Kin  = expf(pk[2 * r]);
      const float Kout = expf(pk[2 * r + 1]);
      if (c == r)                        val = -(((r > 0) ? Kin : 0.0f) + Kout);
      else if (c == r - 1)               val = Kin;            // r >= 1
      else if (c == r + 1 && r < NL - 1) val = Kout;           // last row: Unext = 0
    } else if (bi == 0) {                // d_U gets +f*V
      if (r == c) val = f;
    } else {                             // d_V gets -f*U
      if (r == c) val = -f;
    }
    M[idx] = ((i == j) ? 1.0f : 0.0f) + DT * val;
  }
  if (threadIdx.x < NST) {
    P[threadIdx.x] = (threadIdx.x == 0)  ? 1.0f : 0.0f;   // M^0 e0
    Q[threadIdx.x] = (threadIdx.x == NL) ? 1.0f : 0.0f;   // M^0 e32
  }
}

// Shared WMMA helper: one 16x16 f32 C-tile of D = A(64x64, LDS) * B(64xldb, LDS)
__device__ __forceinline__ v8f junsteak_tile_mm(
    const float* __restrict__ Al, const float* __restrict__ Bl,
    int tm, int tn, int nloc, int hi, int ldb) {
  v8f acc = {};
#pragma unroll
  for (int kb = 0; kb < 16; ++kb) {
    const int k0 = (kb << 2) + (hi << 1);
    v2f a, b;
    // A (16x4 f32): lanes 0-15 -> m=lane, K={k0,k0+1}; lanes 16-31 -> K={k0+2,k0+3}
    a.x = Al[(tm + nloc) * NST + k0];
    a.y = Al[(tm + nloc) * NST + k0 + 1];
    // B (4x16 f32): row k striped across lanes, lane half selects K-pair
    b.x = Bl[(k0) * ldb + tn + nloc];
    b.y = Bl[(k0 + 1) * ldb + tn + nloc];
    acc = __builtin_amdgcn_wmma_f32_16x16x4_f32(
        false, a, false, b, (short)0, acc, false, false);
  }
  return acc;
}

// ---------------- K2: M^2..M^60 via WMMA; harvest P/Q columns -------------------
__global__ void __launch_bounds__(512) junsteak_power_chain(
    const float* __restrict__ Mg, float* __restrict__ Mpow,
    float* __restrict__ P, float* __restrict__ Q) {
  __shared__ float Apow[NST * NST];   // running power M^j (row-major)
  __shared__ float Mc[NST * NST];     // constant M
  const int tid = threadIdx.x;
  for (int idx = tid; idx < NST * NST; idx += 512) {
    const float v = Mg[idx];
    Mc[idx] = v;
    Apow[idx] = v;                    // start at M^1
  }
  __syncthreads();

  const int lane = tid & 31, wave = tid >> 5;
  const int tm = (wave >> 2) << 4, tn = (wave & 3) << 4;
  const int nloc = lane & 15, hi = lane >> 4;

  for (int j = 1; j <= NSUB - 1; ++j) {
    if (tid < NST) {                  // harvest col 0 / col 32 of M^j
      P[j * NST + tid] = Apow[tid * NST + 0];
      Q[j * NST + tid] = Apow[tid * NST + NL];
    }
    v8f acc = junsteak_tile_mm(Mc, Apow, tm, tn, nloc, hi, NST);
    __syncthreads();                  // reads of Apow done before overwrite
#pragma unroll
    for (int r = 0; r < 8; ++r)
      Apow[(tm + r + (hi << 3)) * NST + tn + nloc] = acc[r];
    __syncthreads();
  }
  for (int idx = tid; idx < NST * NST; idx += 512) Mpow[idx] = Apow[idx]; // M^60
}

// ---------------- K2b: W_j = (M^60)^j for j=0..16 via WMMA ----------------------
__global__ void __launch_bounds__(512) junsteak_m60_powers(
    const float* __restrict__ Mpow, float* __restrict__ Wg) {
  __shared__ float Wcur[NST * NST];
  __shared__ float Mc[NST * NST];
  const int tid = threadIdx.x;
  for (int idx = tid; idx < NST * NST; idx += 512) {
    const float v = Mpow[idx];
    Mc[idx] = v;
    Wcur[idx] = v;
    Wg[0 * NST * NST + idx] = ((idx >> 6) == (idx & 63)) ? 1.0f : 0.0f; // W_0 = I
    Wg[1 * NST * NST + idx] = v;                                        // W_1
  }
  __syncthreads();

  const int lane = tid & 31, wave = tid >> 5;
  const int tm = (wave >> 2) << 4, tn = (wave & 3) << 4;
  const int nloc = lane & 15, hi = lane >> 4;

  for (int j = 1; j <= CHUNK - 1; ++j) {          // produce W_2 .. W_16
    v8f acc = junsteak_tile_mm(Mc, Wcur, tm, tn, nloc, hi, NST);
    __syncthreads();
#pragma unroll
    for (int r = 0; r < 8; ++r) {
      const int m = tm + r + (hi << 3);
      Wcur[m * NST + tn + nloc] = acc[r];
      Wg[(size_t)(j + 1) * NST * NST + m * NST + tn + nloc] = acc[r];
    }
    __syncthreads();
  }
}

// ---------------- K3: per-interval forcing vectors c_n (720-way parallel) -------
__global__ void __launch_bounds__(NST) junsteak_forcing(
    const float* __restrict__ TAx, const float* __restrict__ TAy,
    const float* __restrict__ pk,
    const float* __restrict__ P, const float* __restrict__ Q,
    float* __restrict__ Cf, int nT) {
  const int n = blockIdx.x;
  const int i = threadIdx.x;
  const int nnext = (n + 1 >= nT) ? (nT - 1) : (n + 1);  // JAX -1 wrap == last
  const float tax0 = TAx[n], tax1 = TAx[nnext];
  const float tay0 = TAy[n], tay1 = TAy[nnext];
  float acc = 0.0f;
#pragma unroll 4
  for (int k = 0; k < NSUB; ++k) {
    const float aa = (float)k * (1.0f / (float)NSUB);
    const float wx = (1.0f - aa) * tax0 + aa * tax1;
    const float wy = (1.0f - aa) * tay0 + aa * tay1;
    const int jp = (NSUB - 1 - k);
    acc += wx * P[jp * NST + i] + wy * Q[jp * NST + i];
  }
  Cf[n * NST + i] = DT * expf(pk[0]) * acc;
}

// ---------------- K5: chunk-local affine prefixes (parallel over chunks) --------
// l_0 = 0; l_{r+1} = M60 l_r + c_{16g+r}.  Store l_r (r=0..15) and d_g = l_16.
__global__ void __launch_bounds__(NST) junsteak_chunk_local(
    const float* __restrict__ Mpow, const float* __restrict__ Cf,
    float* __restrict__ L, float* __restrict__ Dg, int nT) {
  __shared__ float S[NST];
  const int g = blockIdx.x, i = threadIdx.x;
  float row[NST];
#pragma unroll
  for (int j = 0; j < NST; ++j) row[j] = Mpow[i * NST + j];
  S[i] = 0.0f;
  __syncthreads();
  for (int r = 0; r < CHUNK; ++r) {
    const int n = g * CHUNK + r;
    float acc = 0.0f;
    if (n < nT) {
      L[(size_t)n * NST + i] = S[i];
      acc = Cf[(size_t)n * NST + i];
    }
#pragma unroll
    for (int j = 0; j < NST; ++j) acc += row[j] * S[j];
    __syncthreads();
    S[i] = acc;
    __syncthreads();
  }
  Dg[(size_t)g * NST + i] = S[i];
}

// ---------------- K6: 45-step chunk-boundary scan (the only serial part) --------
// B_0 = 0; B_{g+1} = W_16 B_g + d_g.  Emit B as 64 x BPAD column matrix.
__global__ void __launch_bounds__(NST) junsteak_chunk_scan(
    const float* __restrict__ W16, const float* __restrict__ Dg,
    float* __restrict__ Bmat, int G) {
  __shared__ float S[NST];
  const int i = threadIdx.x;
  float row[NST];
#pragma unroll
  for (int j = 0; j < NST; ++j) row[j] = W16[i * NST + j];
  for (int c = G; c < BPAD; ++c) Bmat[i * BPAD + c] = 0.0f;   // pad tiles
  S[i] = 0.0f;
  __syncthreads();
  for (int g = 0; g < G; ++g) {
    Bmat[i * BPAD + g] = S[i];        // B_g, column g
    float acc = Dg[(size_t)g * NST + i];
#pragma unroll
    for (int j = 0; j < NST; ++j) acc += row[j] * S[j];
    __syncthreads();
    S[i] = acc;
    __syncthreads();
  }
}

// ---------------- K7: WMMA fix-up X_r = W_r * B (64xBPAD) + output --------------
__global__ void __launch_bounds__(512) junsteak_fixup(
    const float* __restrict__ Wg, const float* __restrict__ Bmat,
    const float* __restrict__ L, float* __restrict__ out, int nT, int G) {
  __shared__ float Wl[NST * NST];
  __shared__ float Bl[NST * BPAD];
  __shared__ float Xl[NST * BPAD];
  const int r = blockIdx.x;           // substep-in-chunk 0..15
  const int tid = threadIdx.x;
  for (int idx = tid; idx < NST * NST; idx += 512)
    Wl[idx] = Wg[(size_t)r * NST * NST + idx];
  for (int idx = tid; idx < NST * BPAD; idx += 512) Bl[idx] = Bmat[idx];
  __syncthreads();

  const int lane = tid & 31, wave = tid >> 5;
  const int nloc = lane & 15, hi = lane >> 4;
  if (wave < 12) {                    // 4 row-tiles x 3 col-tiles, EXEC all-1s
    const int tm = (wave / 3) << 4, tn = (wave % 3) << 4;
    v8f acc = junsteak_tile_mm(Wl, Bl, tm, tn, nloc, hi, BPAD);
#pragma unroll
    for (int rr = 0; rr < 8; ++rr)
      Xl[(tm + rr + (hi << 3)) * BPAD + tn + nloc] = acc[rr];
  }
  __syncthreads();

  float* __restrict__ outU = out;
  float* __restrict__ outV = out + (size_t)nT * NL;
  for (int idx = tid; idx < G * NST; idx += 512) {
    const int g = idx >> 6, i = idx & 63;
    const int n = g * CHUNK + r;
    if (n < nT) {
      const float v = Xl[i * BPAD + g] + L[(size_t)n * NST + i];
      if (i < NL) outU[(size_t)n * NL + i] = v;
      else        outV[(size_t)n * NL + (i - NL)] = v;
    }
  }
}

extern "C" void kernel_launch(void* const* d_in, const int* in_sizes, int n_in,
                              void* d_out, int out_size, void* d_ws, size_t ws_size,
                              hipStream_t stream) {
  const float* pk  = (const float*)d_in[0];
  const float* TAx = (const float*)d_in[1];
  const float* TAy = (const float*)d_in[2];
  const float* fc  = (const float*)d_in[3];
  const int nT = in_sizes[1];                 // 720 forcing intervals
  const int G  = (nT + CHUNK - 1) / CHUNK;    // 45 chunks (<= BPAD)

  float* ws   = (float*)d_ws;
  float* M    = ws;                           // 64*64
  float* Mpow = M + NST * NST;                // 64*64      (M^60)
  float* P    = Mpow + NST * NST;             // NSUB*64    (col0 of M^j)
  float* Q    = P + NSUB * NST;               // NSUB*64    (col32 of M^j)
  float* Cf   = Q + NSUB * NST;               // nT*64      (forcing vectors)
  float* Wg   = Cf + (size_t)nT * NST;        // 17*64*64   ((M^60)^j, j=0..16)
  float* L    = Wg + (size_t)(CHUNK + 1) * NST * NST;  // nT*64 (local prefixes)
  float* Dg   = L + (size_t)nT * NST;         // G*64       (chunk increments)
  float* Bmat = Dg + (size_t)G * NST;         // 64*BPAD    (boundary states)

  junsteak_build_M<<<1, 256, 0, stream>>>(pk, fc, M, P, Q);
  junsteak_power_chain<<<1, 512, 0, stream>>>(M, Mpow, P, Q);
  junsteak_m60_powers<<<1, 512, 0, stream>>>(Mpow, Wg);
  junsteak_forcing<<<nT, NST, 0, stream>>>(TAx, TAy, pk, P, Q, Cf, nT);
  junsteak_chunk_local<<<G, NST, 0, stream>>>(Mpow, Cf, L, Dg, nT);
  junsteak_chunk_scan<<<1, NST, 0, stream>>>(Wg + (size_t)CHUNK * NST * NST, Dg, Bmat, G);
  junsteak_fixup<<<CHUNK, 512, 0, stream>>>(Wg, Bmat, L, (float*)d_out, nT, G);
}